// HyperbolicMuRPRel_69260642615718
// MI455X (gfx1250) — compile-verified
//
#include <hip/hip_runtime.h>
#include <math.h>

typedef float v2f __attribute__((ext_vector_type(2)));
typedef float v8f __attribute__((ext_vector_type(8)));

#define DDIM 256
#define NB   1024
#define NR   512
#define CC     0.01f
#define SQRT_C 0.1f
#define EPSF   1e-6f

__device__ __forceinline__ float block_reduce_sum(float v, float* sh) {
  int tid = threadIdx.x;
  sh[tid] = v;
  __syncthreads();
  for (int s = 128; s > 0; s >>= 1) {
    if (tid < s) sh[tid] += sh[tid + s];
    __syncthreads();
  }
  float r = sh[0];
  __syncthreads();
  return r;
}

// ---- Kernel 1: gather entity rows + log_map_zero ------------------------
// grid (1024, 2): y=0 -> subject rows into Ls, y=1 -> object rows into Lo
__global__ void logmap_gather(const float* __restrict__ ent,
                              const int*   __restrict__ trip,
                              float* __restrict__ Ls,
                              float* __restrict__ Lo) {
  __shared__ float sh[DDIM];
  int b = blockIdx.x;
  int which = blockIdx.y;
  int e = trip[b * 3 + (which ? 2 : 0)];
  float x = ent[(long long)e * DDIM + threadIdx.x];
  float n2 = block_reduce_sum(x * x, sh);
  float norm = fmaxf(sqrtf(n2), EPSF);
  float arg = fminf(SQRT_C * norm, 1.0f - 1e-5f);
  float scale = atanhf(arg) / (SQRT_C * norm);
  float* out = which ? Lo : Ls;
  out[b * DDIM + threadIdx.x] = scale * x;
}

// ---- Kernel 2: tan = L @ W via V_WMMA_F32_16X16X4_F32 -------------------
// grid (128, 2), block 256: 8 waves/block, 1 wave = one 16x16 output tile.
__global__ void gemm_tan(const float* __restrict__ Ls,
                         const float* __restrict__ Lo,
                         const float* __restrict__ Ws,
                         const float* __restrict__ Wo,
                         float* __restrict__ st,
                         float* __restrict__ ot) {
  const float* A = blockIdx.y ? Lo : Ls;
  const float* W = blockIdx.y ? Wo : Ws;
  float* Cout    = blockIdx.y ? ot : st;

  int wave   = blockIdx.x * (blockDim.x >> 5) + (threadIdx.x >> 5);
  int tile_m = wave >> 4;        // 16 N-tiles (N = 256)
  int tile_n = wave & 15;
  int lane   = threadIdx.x & 31;
  int row    = lane & 15;                 // A row within tile
  int koff   = (lane >> 4) << 1;          // K pair: 0 for lanes 0-15, 2 for 16-31
  int col    = lane & 15;                 // B col within tile

  v8f acc = {};
  const float* Arow = A + (tile_m * 16 + row) * DDIM;
  for (int k0 = 0; k0 < DDIM; k0 += 4) {
    v2f a = *(const v2f*)(Arow + k0 + koff);        // A[row][k0+koff .. +1]
    v2f b;
    b.x = W[(k0 + koff)     * DDIM + tile_n * 16 + col];
    b.y = W[(k0 + koff + 1) * DDIM + tile_n * 16 + col];
    acc = __builtin_amdgcn_wmma_f32_16x16x4_f32(false, a, false, b,
                                                (short)0, acc, false, false);
  }
  int mbase = tile_m * 16 + ((lane >> 4) << 3);     // +8 for upper lane half
  int cg    = tile_n * 16 + col;
#pragma unroll
  for (int i = 0; i < 8; ++i)
    Cout[(mbase + i) * DDIM + cg] = acc[i];
}

// ---- Kernel 3: exp_map both rows + mobius_add(-s_h, o_h) -> query -------
__global__ void combine_query(const float* __restrict__ st,
                              const float* __restrict__ ot,
                              float* __restrict__ q,
                              float* __restrict__ qn2) {
  __shared__ float sh[DDIM];
  int b = blockIdx.x, tid = threadIdx.x;
  float s = st[b * DDIM + tid];
  float o = ot[b * DDIM + tid];
  float ns2 = block_reduce_sum(s * s, sh);
  float no2 = block_reduce_sum(o * o, sh);
  float dot = block_reduce_sum(s * o, sh);

  float ns = fmaxf(sqrtf(ns2), EPSF);
  float no = fmaxf(sqrtf(no2), EPSF);
  float fs = tanhf(SQRT_C * ns) / (SQRT_C * ns);    // exp_map scale for s
  float fo = tanhf(SQRT_C * no) / (SQRT_C * no);    // exp_map scale for o

  // mobius_add with x = -s_h, y = o_h
  float x2 = fs * fs * ns2;
  float y2 = fo * fo * no2;
  float xy = -fs * fo * dot;
  float t = 2.0f * CC * xy;
  float alpha = 1.0f + t + CC * y2;
  float beta  = 1.0f - CC * x2;
  float den   = 1.0f + t + CC * CC * x2 * y2 + EPSF;

  float xi = -fs * s, yi = fo * o;
  q[b * DDIM + tid] = (alpha * xi + beta * yi) / den;
  if (tid == 0)
    qn2[b] = (alpha * alpha * x2 + 2.0f * alpha * beta * xy + beta * beta * y2)
             / (den * den);
}

// ---- Kernel 4: rel_h = exp_map_zero(rel_embedding), plus row norms ------
__global__ void rel_exp(const float* __restrict__ rel,
                        float* __restrict__ relh,
                        float* __restrict__ y2r) {
  __shared__ float sh[DDIM];
  int r = blockIdx.x, tid = threadIdx.x;
  float v = rel[r * DDIM + tid];
  float n2 = block_reduce_sum(v * v, sh);
  float norm = fmaxf(sqrtf(n2), EPSF);
  float f = tanhf(SQRT_C * norm) / (SQRT_C * norm);
  relh[r * DDIM + tid] = f * v;
  if (tid == 0) y2r[r] = f * f * n2;
}

// ---- Kernel 5: G = query @ rel_h^T (WMMA) + fused mobius-distance score -
// grid 256, block 256: 2048 waves = 64 x 32 tiles of the 1024x512 output.
__global__ void score_gemm(const float* __restrict__ q,
                           const float* __restrict__ relh,
                           const float* __restrict__ qn2,
                           const float* __restrict__ y2r,
                           const float* __restrict__ bias,
                           float* __restrict__ out) {
  int wave   = blockIdx.x * (blockDim.x >> 5) + (threadIdx.x >> 5);
  int tile_m = wave >> 5;        // 32 N-tiles (N = 512)
  int tile_n = wave & 31;
  int lane   = threadIdx.x & 31;
  int row    = lane & 15;
  int koff   = (lane >> 4) << 1;
  int col    = lane & 15;

  v8f acc = {};
  const float* Arow = q    + (tile_m * 16 + row) * DDIM;
  const float* Brow = relh + (tile_n * 16 + col) * DDIM;   // B[k][n] = relh[n][k]
  for (int k0 = 0; k0 < DDIM; k0 += 4) {
    v2f a = *(const v2f*)(Arow + k0 + koff);
    v2f b = *(const v2f*)(Brow + k0 + koff);
    acc = __builtin_amdgcn_wmma_f32_16x16x4_f32(false, a, false, b,
                                                (short)0, acc, false, false);
  }

  int mbase = tile_m * 16 + ((lane >> 4) << 3);
  int cg    = tile_n * 16 + col;
  float y2 = y2r[cg];
  float bv = bias[cg];
#pragma unroll
  for (int i = 0; i < 8; ++i) {
    float G  = acc[i];                 // <query_b, relh_r>
    float x2 = qn2[mbase + i];
    float xy = -G;                     // x = -query
    float t = 2.0f * CC * xy;
    float alpha = 1.0f + t + CC * y2;
    float beta  = 1.0f - CC * x2;
    float den   = 1.0f + t + CC * CC * x2 * y2 + EPSF;
    float dist  = (alpha * alpha * x2 + 2.0f * alpha * beta * xy
                   + beta * beta * y2) / (den * den);
    out[(mbase + i) * NR + cg] = -dist + bv;
  }
}

extern "C" void kernel_launch(void* const* d_in, const int* in_sizes, int n_in,
                              void* d_out, int out_size, void* d_ws, size_t ws_size,
                              hipStream_t stream) {
  const float* ent  = (const float*)d_in[0];   // (20000, 256)
  const float* rel  = (const float*)d_in[1];   // (512, 256)
  const int*   trip = (const int*)  d_in[2];   // (1024, 3)
  const float* Ws   = (const float*)d_in[3];   // (256, 256)
  const float* Wo   = (const float*)d_in[4];   // (256, 256)
  const float* bias = (const float*)d_in[5];   // (512,)
  float* out = (float*)d_out;                  // (1024, 512)

  float* ws   = (float*)d_ws;
  float* Ls   = ws;                // 1024*256
  float* Lo   = Ls + NB * DDIM;    // 1024*256
  float* st   = Lo + NB * DDIM;    // 1024*256
  float* ot   = st + NB * DDIM;    // 1024*256
  float* q    = ot + NB * DDIM;    // 1024*256
  float* relh = q  + NB * DDIM;    // 512*256
  float* qn2  = relh + NR * DDIM;  // 1024
  float* y2r  = qn2 + NB;          // 512

  logmap_gather<<<dim3(NB, 2), DDIM, 0, stream>>>(ent, trip, Ls, Lo);
  gemm_tan<<<dim3(128, 2), 256, 0, stream>>>(Ls, Lo, Ws, Wo, st, ot);
  combine_query<<<NB, DDIM, 0, stream>>>(st, ot, q, qn2);
  rel_exp<<<NR, DDIM, 0, stream>>>(rel, relh, y2r);
  score_gemm<<<256, 256, 0, stream>>>(q, relh, qn2, y2r, bias, out);
}